// GATLayer_76802605187271
// MI455X (gfx1250) — compile-verified
//
#include <hip/hip_runtime.h>
#include <stdint.h>

typedef __attribute__((ext_vector_type(2))) float v2f;
typedef __attribute__((ext_vector_type(8))) float v8f;
typedef __attribute__((ext_vector_type(4))) unsigned int v4u;
typedef __attribute__((ext_vector_type(8))) int v8i;
typedef __attribute__((ext_vector_type(4))) int v4i;

#define FIN   256
#define FOUT  128
#define NNODE 2048
#define NB    8
#define NROW  (NB * NNODE)      // 16384 rows of h / Wh
#define ALPHA 0.2f

// ---------------- Kernel A: Wh = h @ W  (WMMA f32 16x16x4), plus
//                  ei = Wh@a1, ej = Wh@a2 accumulated with atomics ----------
#define KC            64        // K-chunk staged in LDS
#define LDA           68        // 64 + 4 pad floats -> bank-conflict-free
#define ROWS_PER_BLK  64        // rows of Wh per block (256 thr / 8 waves)

__global__ __launch_bounds__(256)
void gat_gemm_wmma(const float* __restrict__ h,
                   const float* __restrict__ W,
                   const float* __restrict__ avec,
                   float* __restrict__ Wh,
                   float* __restrict__ ei,
                   float* __restrict__ ej)
{
    __shared__ __attribute__((aligned(16))) float Ash[ROWS_PER_BLK * LDA]; // 64 x 68
    __shared__ __attribute__((aligned(16))) float Bsh[FOUT * LDA];         // 128 x 68 (W^T chunk)

    const int tid   = threadIdx.x;
    const int lane  = tid & 31;
    const int wv    = tid >> 5;
    const int g0    = blockIdx.x * ROWS_PER_BLK;

    const int rt    = wv & 3;              // row tile: rows rt*16 .. +15
    const int cbase = (wv >> 2) * 64;      // col half: 0 or 64
    const int nn    = lane & 15;           // N (or M) within tile
    const int kh    = (lane >> 4) * 2;     // K offset for upper half-wave
    const int hi8   = (lane >> 4) << 3;    // +8 rows for lanes 16..31 (C/D layout)

#if __has_builtin(__builtin_amdgcn_tensor_load_to_lds) && __has_builtin(__builtin_amdgcn_s_wait_tensorcnt)
    // LDS byte offset of Ash: flat LDS-aperture address truncated to 32 bits
    const uint32_t ashBase = (uint32_t)(uintptr_t)(void*)&Ash[0];
#endif

    v8f acc[4] = {};                       // 4 col tiles of 16x16 f32

    #pragma unroll 1
    for (int kc = 0; kc < FIN; kc += KC) {
        __syncthreads();

#if __has_builtin(__builtin_amdgcn_tensor_load_to_lds) && __has_builtin(__builtin_amdgcn_s_wait_tensorcnt)
        // ---- TDM: DMA h tile (64 rows x 64 k, fp32) straight into LDS with
        //      4-DWORD padding after every 64 DWORDs -> rows land at stride LDA.
        if (wv == 0) {
            const uint64_t ga = (uint64_t)(uintptr_t)(h + (size_t)g0 * FIN + kc);
            // D# group0: count=1 | lds_addr | global_addr | type=2
            v4u d0;
            d0[0] = 1u;                                     // count=1, user mode
            d0[1] = ashBase;                                // lds_addr (bytes)
            d0[2] = (uint32_t)ga;                           // global_addr[31:0]
            d0[3] = ((uint32_t)(ga >> 32) & 0x01ffffffu)    // global_addr[56:32]
                    | 0x80000000u;                          // type=2 ("image")
            // D# group1
            v8i d1;
            d1[0] = (int)((2u << 16)       // data_size = 4B
                        | (1u << 20)       // pad_enable
                        | (5u << 22)       // pad_interval: 64 DWORDs
                        | (3u << 25));     // pad_amount: 4 DWORDs
            d1[1] = (int)(((uint32_t)FIN & 0xffffu) << 16);           // tensor_dim0[15:0]
            d1[2] = (int)(((uint32_t)FIN >> 16)
                        | (((uint32_t)NROW & 0xffffu) << 16));        // dim0 hi | dim1 lo
            d1[3] = (int)(((uint32_t)NROW >> 16) | (64u << 16));      // dim1 hi | tile_dim0=64
            d1[4] = 64;                                               // tile_dim1=64, tile_dim2=0
            d1[5] = FIN;                                              // tensor_dim0_stride = 256
            d1[6] = 0;
            d1[7] = 0;
            v4i zz4 = {0, 0, 0, 0};
            v8i zz8 = {0, 0, 0, 0, 0, 0, 0, 0};
            __builtin_amdgcn_tensor_load_to_lds(d0, d1, zz4, zz4, zz8, 0); // tensor_load_to_lds
        }
#else
        // ---- fallback: manual stage A chunk (1024 float4, 4/thread) ----
        {
            const float4* hg = (const float4*)h;
            #pragma unroll
            for (int i = 0; i < 4; ++i) {
                int e4 = tid + i * 256;            // 0..1023
                int r  = e4 >> 4;                  // 16 float4 per row
                int k4 = e4 & 15;
                float4 v = hg[(((size_t)(g0 + r) * FIN + kc) >> 2) + k4];
                float* dst = &Ash[r * LDA + k4 * 4];
                dst[0] = v.x; dst[1] = v.y; dst[2] = v.z; dst[3] = v.w;
            }
        }
#endif

        // ---- stage W chunk transposed: Bsh[o][k] = W[kc+k][o] (all threads) ----
        {
            const float4* wg = (const float4*)W;
            #pragma unroll
            for (int i = 0; i < 8; ++i) {
                int e4 = tid + i * 256;            // 0..2047
                int k  = e4 >> 5;                  // 32 float4 per k-row
                int o  = (e4 & 31) * 4;
                float4 v = wg[(((size_t)(kc + k) * FOUT) >> 2) + (o >> 2)];
                Bsh[(o + 0) * LDA + k] = v.x;
                Bsh[(o + 1) * LDA + k] = v.y;
                Bsh[(o + 2) * LDA + k] = v.z;
                Bsh[(o + 3) * LDA + k] = v.w;
            }
        }

#if __has_builtin(__builtin_amdgcn_tensor_load_to_lds) && __has_builtin(__builtin_amdgcn_s_wait_tensorcnt)
        if (wv == 0) __builtin_amdgcn_s_wait_tensorcnt(0);  // DMA landed before barrier
#endif
        __syncthreads();

        if (kc + KC < FIN)
            __builtin_prefetch(&W[(size_t)(kc + KC) * FOUT + tid], 0, 1); // global_prefetch_b8

        const float* Arow = &Ash[(rt * 16 + nn) * LDA + kh];
        const float* B0   = &Bsh[(cbase +  0 + nn) * LDA + kh];
        const float* B1   = &Bsh[(cbase + 16 + nn) * LDA + kh];
        const float* B2   = &Bsh[(cbase + 32 + nn) * LDA + kh];
        const float* B3   = &Bsh[(cbase + 48 + nn) * LDA + kh];

        #pragma unroll
        for (int k = 0; k < KC; k += 4) {
            v2f a  = *(const v2f*)(Arow + k);
            v2f b0 = *(const v2f*)(B0 + k);
            acc[0] = __builtin_amdgcn_wmma_f32_16x16x4_f32(false, a, false, b0,
                                                           (short)0, acc[0], false, false);
            v2f b1 = *(const v2f*)(B1 + k);
            acc[1] = __builtin_amdgcn_wmma_f32_16x16x4_f32(false, a, false, b1,
                                                           (short)0, acc[1], false, false);
            v2f b2 = *(const v2f*)(B2 + k);
            acc[2] = __builtin_amdgcn_wmma_f32_16x16x4_f32(false, a, false, b2,
                                                           (short)0, acc[2], false, false);
            v2f b3 = *(const v2f*)(B3 + k);
            acc[3] = __builtin_amdgcn_wmma_f32_16x16x4_f32(false, a, false, b3,
                                                           (short)0, acc[3], false, false);
        }
    }

    // ---- epilogue: store Wh tiles; fold in ei = Wh@a1, ej = Wh@a2 ----
    #pragma unroll
    for (int ct = 0; ct < 4; ++ct) {
        const int   c0  = cbase + ct * 16;
        const float a1c = avec[c0 + nn];
        const float a2c = avec[FOUT + c0 + nn];
        #pragma unroll
        for (int r = 0; r < 8; ++r) {
            const float v   = acc[ct][r];
            const int   row = g0 + rt * 16 + r + hi8;     // C/D layout: M = r (+8 upper half)
            Wh[(size_t)row * FOUT + c0 + nn] = v;

            float p1 = v * a1c;
            float p2 = v * a2c;
            #pragma unroll
            for (int off = 1; off < 16; off <<= 1) {      // reduce across the 16 N-lanes
                p1 += __shfl_xor(p1, off, 32);
                p2 += __shfl_xor(p2, off, 32);
            }
            if (nn == 0) {
                atomicAdd(&ei[row], p1);
                atomicAdd(&ej[row], p2);
            }
        }
    }
}

// ---------------- Kernel B: column sums of softmax(lrelu(ei_i + ej_j)) -----
// w[b,j] = (1/N) * sum_i softmax_j(...)[j].  ej[b,:] lives in LDS; per-thread
// register accumulators over 8 fixed j columns; row max = lrelu(ei + max(ej)).
__global__ __launch_bounds__(256)
void gat_attn_colsum(const float* __restrict__ ei,
                     const float* __restrict__ ej,
                     float* __restrict__ wcol)
{
    __shared__ float ejs[NNODE];   // 8 KB
    __shared__ float red[8];
    __shared__ float sbc;
    __shared__ float mbc;

    const int b    = blockIdx.x;
    const int i0   = blockIdx.y * 128;
    const int tid  = threadIdx.x;
    const int lane = tid & 31;
    const int wv   = tid >> 5;

    float lmax = -3.402823466e38f;
    #pragma unroll
    for (int k = 0; k < 8; ++k) {
        int   j = tid + k * 256;
        float v = ej[b * NNODE + j];
        ejs[j]  = v;
        lmax    = fmaxf(lmax, v);
    }
    #pragma unroll
    for (int off = 16; off; off >>= 1) lmax = fmaxf(lmax, __shfl_xor(lmax, off, 32));
    if (lane == 0) red[wv] = lmax;
    __syncthreads();
    if (tid == 0) {
        float m = red[0];
        for (int w = 1; w < 8; ++w) m = fmaxf(m, red[w]);
        mbc = m;
    }
    __syncthreads();
    const float maxej = mbc;

    float wacc[8] = {0.f, 0.f, 0.f, 0.f, 0.f, 0.f, 0.f, 0.f};

    for (int ii = 0; ii < 128; ++ii) {
        const float eiv  = ei[b * NNODE + i0 + ii];
        float zmax = eiv + maxej;
        zmax = (zmax >= 0.f) ? zmax : ALPHA * zmax;   // monotone lrelu -> true row max

        float p[8];
        float sp = 0.f;
        #pragma unroll
        for (int k = 0; k < 8; ++k) {
            float z = eiv + ejs[tid + k * 256];
            z = (z >= 0.f) ? z : ALPHA * z;
            float e = __expf(z - zmax);               // v_exp_f32
            p[k] = e;
            sp  += e;
        }
        #pragma unroll
        for (int off = 16; off; off >>= 1) sp += __shfl_xor(sp, off, 32);
        if (lane == 0) red[wv] = sp;
        __syncthreads();
        if (tid == 0) {
            float s = 0.f;
            for (int w = 0; w < 8; ++w) s += red[w];
            sbc = s;
        }
        __syncthreads();
        const float rinv = __builtin_amdgcn_rcpf(sbc);
        #pragma unroll
        for (int k = 0; k < 8; ++k) wacc[k] += p[k] * rinv;
        __syncthreads();
    }

    const float sc = 1.0f / (float)NNODE;             // the mean over i
    #pragma unroll
    for (int k = 0; k < 8; ++k)
        atomicAdd(&wcol[b * NNODE + tid + k * 256], wacc[k] * sc);
}

// ---------------- Kernel C: out[b,:] = sum_j w[b,j] * Wh[b,j,:] ------------
__global__ __launch_bounds__(128)
void gat_out(const float* __restrict__ Wh,
             const float* __restrict__ wcol,
             float* __restrict__ out)
{
    const int b = blockIdx.x;
    const int o = threadIdx.x;
    const float* whb = Wh + (size_t)b * NNODE * FOUT;
    const float* wb  = wcol + b * NNODE;
    float acc = 0.f;
    #pragma unroll 8
    for (int j = 0; j < NNODE; ++j)
        acc = fmaf(wb[j], whb[(size_t)j * FOUT + o], acc);
    out[b * FOUT + o] = acc;
}

// ---------------------------------------------------------------------------
extern "C" void kernel_launch(void* const* d_in, const int* in_sizes, int n_in,
                              void* d_out, int out_size, void* d_ws, size_t ws_size,
                              hipStream_t stream) {
    (void)in_sizes; (void)n_in; (void)out_size; (void)ws_size;

    const float* h = (const float*)d_in[0];   // (8,2048,256)
    const float* W = (const float*)d_in[1];   // (256,128)
    const float* a = (const float*)d_in[2];   // (256,)
    float* out = (float*)d_out;               // (8,128)

    float* Wh   = (float*)d_ws;                           // 16384*128 f32 (8 MB)
    float* ei   = Wh + (size_t)NROW * FOUT;               // 16384 f32
    float* ej   = ei + NROW;                              // 16384 f32
    float* wcol = ej + NROW;                              // 16384 f32

    (void)hipMemsetAsync(ei, 0, (size_t)3 * NROW * sizeof(float), stream);

    gat_gemm_wmma<<<NROW / ROWS_PER_BLK, 256, 0, stream>>>(h, W, a, Wh, ei, ej);
    gat_attn_colsum<<<dim3(NB, NNODE / 128), 256, 0, stream>>>(ei, ej, wcol);
    gat_out<<<NB, FOUT, 0, stream>>>(Wh, wcol, out);
}